// Encoder_57062935494680
// MI455X (gfx1250) — compile-verified
//
#include <hip/hip_runtime.h>
#include <math.h>

// ---------------- problem constants ----------------
constexpr int N_NODES = 8192;
constexpr int E_EDGES = 262144;
constexpr int IN_F    = 3000;
constexpr int HID_F   = 256;
constexpr int OUT_F   = 64;

// GEMM epilogue flags
#define GF_BIAS 1
#define GF_ELU  2

// K-chunk staged through LDS per block
#define KC 64

typedef float v2f __attribute__((ext_vector_type(2)));
typedef float v8f __attribute__((ext_vector_type(8)));

__device__ __forceinline__ float elu_f(float x) { return x > 0.f ? x : (expf(x) - 1.f); }

// order-preserving float<->uint mapping for atomic max on signed floats
__device__ __forceinline__ unsigned f2ord(float f) {
    unsigned u = __float_as_uint(f);
    return (u & 0x80000000u) ? ~u : (u | 0x80000000u);
}
__device__ __forceinline__ float ord2f(unsigned u) {
    return __uint_as_float((u & 0x80000000u) ? (u ^ 0x80000000u) : ~u);
}

// Async global->LDS fill of one full KC x 64 B-chunk (16 KB) using
// GLOBAL_LOAD_ASYNC_TO_LDS_B128 (ASYNCcnt-tracked, no VGPR round trip).
// 256 threads x 4 ops x 16 B = 16 KB. Requires 16B-aligned rows
// (ldb % 4 == 0, true for all call sites) and a fully in-range tile.
__device__ __forceinline__ void async_fill_b128(float* __restrict__ dst,
                                                const float* __restrict__ B,
                                                int ldb, int k0, int c0, int tid)
{
#pragma unroll
    for (int p = 0; p < 4; ++p) {
        const int q  = tid + p * 256;          // float4 index within the tile
        const int kk = q >> 4;                 // K row inside chunk
        const int n  = (q & 15) << 2;          // 4-float group inside the 64 cols
        const float* gp = B + (size_t)(k0 + kk) * ldb + c0 + n;
        const unsigned lds = (unsigned)(size_t)(dst + (q << 2));
        const unsigned long long ga = (unsigned long long)(size_t)gp;
        asm volatile("global_load_async_to_lds_b128 %0, %1, off"
                     :: "v"(lds), "v"(ga) : "memory");
    }
}

__device__ __forceinline__ void wait_asynccnt0()
{
    asm volatile("s_wait_asynccnt 0x0" ::: "memory");
}

// ---------------------------------------------------------------------------
// fp32 WMMA GEMM: C[M,N] = A[M,K] @ B[K,N] (+bias[col]) (elu)
// Row-major. Requires M % 128 == 0, K % 4 == 0. N edge handled by the
// synchronous fallback fill and the store loop only.
// Block = 256 threads = 8 waves; block tile = 128 rows x 64 cols.
// Each wave: 16 rows x 64 cols (4 accumulators, A fragment reused 4x).
// B tile (KC x 64) staged in LDS, double-buffered via async-to-LDS copies
// so the fill of chunk i+1 overlaps the WMMA work on chunk i.
// V_WMMA_F32_16X16X4_F32 operand layout per CDNA5 ISA 7.12.2:
//   A 16x4 : lane<16 -> M=lane, {K0,K1}; lane>=16 -> M=lane-16, {K2,K3}
//   B 4x16 : lane<16 -> N=lane, {K0,K1}; lane>=16 -> N=lane-16, {K2,K3}
//   C 16x16: vgpr v, lane l -> M = v + 8*(l>=16), N = l%16
// ---------------------------------------------------------------------------
__global__ void wmma_gemm_f32(const float* __restrict__ A, const float* __restrict__ B,
                              const float* __restrict__ bias, float* __restrict__ C,
                              int M, int K, int Ncols, int flags)
{
    __shared__ __align__(16) float Bs[2][KC * 64];

    const int tid  = threadIdx.x;
    const int wave = tid >> 5;
    const int lane = tid & 31;
    const int r0   = blockIdx.x * 128 + wave * 16;
    const int c0   = blockIdx.y * 64;
    const int lm   = lane & 15;
    const int kh   = (lane >> 4) << 1;   // 0 or 2

    v8f acc[4];
#pragma unroll
    for (int t = 0; t < 4; ++t)
#pragma unroll
        for (int i = 0; i < 8; ++i) acc[t][i] = 0.f;

    const float* Ap = A + (size_t)(r0 + lm) * K;

    // guard-free WMMA sweep over one staged chunk
    auto compute = [&](const float* __restrict__ Bsc, int k0, int kc) {
        __builtin_prefetch(Ap + k0 + KC, 0, 3);   // speculative next-A prefetch
        for (int kk = 0; kk < kc; kk += 4) {
            const int kb = kk + kh;
            v2f a;
            a.x = Ap[k0 + kb];
            a.y = Ap[k0 + kb + 1];
            const float* Bl0 = Bsc + kb * 64;
            const float* Bl1 = Bl0 + 64;
#pragma unroll
            for (int t = 0; t < 4; ++t) {
                v2f b;
                b.x = Bl0[t * 16 + lm];
                b.y = Bl1[t * 16 + lm];
                acc[t] = __builtin_amdgcn_wmma_f32_16x16x4_f32(
                    false, a, false, b, (short)0, acc[t], false, false);
            }
        }
    };

    const int nfull = K / KC;
    const int ktail = K - nfull * KC;

    if (c0 + 64 <= Ncols && nfull > 0) {
        // ---------- async double-buffered path (tile fully in-range) ----------
        async_fill_b128(&Bs[0][0], B, Ncols, 0, c0, tid);
        for (int i = 0; i < nfull; ++i) {
            const float* cur = &Bs[i & 1][0];
            wait_asynccnt0();          // my slice of chunk i landed
            __syncthreads();           // everyone waited first -> all of chunk i landed,
                                       // and all waves are done reading the other buffer
            if (i + 1 < nfull)
                async_fill_b128(&Bs[(i + 1) & 1][0], B, Ncols, (i + 1) * KC, c0, tid);
            compute(cur, i * KC, KC);  // overlaps the async fill of chunk i+1
        }
        if (ktail) {
            float* tb = &Bs[nfull & 1][0];
            __syncthreads();
            for (int idx = tid; idx < ktail * 64; idx += 256)
                tb[idx] = B[(size_t)(nfull * KC + (idx >> 6)) * Ncols + c0 + (idx & 63)];
            __syncthreads();
            compute(tb, nfull * KC, ktail);
        }
    } else {
        // ---------- synchronous guarded fallback (N-edge blocks) ----------
        for (int k0 = 0; k0 < K; k0 += KC) {
            const int kc = (K - k0 < KC) ? (K - k0) : KC;
            __syncthreads();
            for (int idx = tid; idx < kc * 64; idx += 256) {
                const int kk = idx >> 6;
                const int n  = c0 + (idx & 63);
                Bs[0][idx] = (n < Ncols) ? B[(size_t)(k0 + kk) * Ncols + n] : 0.f;
            }
            __syncthreads();
            compute(&Bs[0][0], k0, kc);
        }
    }

    const int rbase = r0 + ((lane >> 4) << 3);
#pragma unroll
    for (int v = 0; v < 8; ++v) {
        const int row = rbase + v;
#pragma unroll
        for (int t = 0; t < 4; ++t) {
            const int col = c0 + t * 16 + lm;
            if (col < Ncols) {
                float x = acc[t][v];
                if (flags & GF_BIAS) x += bias[col];
                if (flags & GF_ELU)  x = elu_f(x);
                C[(size_t)row * Ncols + col] = x;
            }
        }
    }
}

// ---------------- attention dot products: a_src = h@att_src, a_dst = h@att_dst
__global__ void att_dots(const float* __restrict__ h, const float* __restrict__ att_s,
                         const float* __restrict__ att_d, float* __restrict__ as,
                         float* __restrict__ ad)
{
    __shared__ float s1[256], s2[256];
    const int n = blockIdx.x, t = threadIdx.x;
    const float hv = h[(size_t)n * HID_F + t];
    s1[t] = hv * att_s[t];
    s2[t] = hv * att_d[t];
    __syncthreads();
    for (int k = 128; k > 0; k >>= 1) {
        if (t < k) { s1[t] += s1[t + k]; s2[t] += s2[t + k]; }
        __syncthreads();
    }
    if (t == 0) { as[n] = s1[0]; ad[n] = s2[0]; }
}

// ---------------- per-graph init: m=-inf (ordered), z=0, h1=0
__global__ void init_graph_bufs(unsigned* __restrict__ mb, float* __restrict__ zb,
                                float* __restrict__ h1)
{
    const int i = blockIdx.x * blockDim.x + threadIdx.x;   // grid covers N*HID
    if (i < N_NODES) { mb[i] = f2ord(-INFINITY); zb[i] = 0.f; }
    if (i < N_NODES * HID_F) h1[i] = 0.f;
}

// ---------------- edge phase ----------------
__global__ void edge_logits(const int* __restrict__ src, const int* __restrict__ dst,
                            const float* __restrict__ as, const float* __restrict__ ad,
                            float* __restrict__ ebuf, unsigned* __restrict__ mb)
{
    const int i = blockIdx.x * blockDim.x + threadIdx.x;
    if (i >= E_EDGES) return;
    float e = as[src[i]] + ad[dst[i]];
    e = e > 0.f ? e : 0.2f * e;                 // leaky_relu slope 0.2
    ebuf[i] = e;
    atomicMax(mb + dst[i], f2ord(e));
}

__global__ void edge_softmax_num(const int* __restrict__ dst, const unsigned* __restrict__ mb,
                                 float* __restrict__ ebuf, float* __restrict__ zb)
{
    const int i = blockIdx.x * blockDim.x + threadIdx.x;
    if (i >= E_EDGES) return;
    const int d = dst[i];
    const float p = expf(ebuf[i] - ord2f(mb[d]));
    ebuf[i] = p;
    atomicAdd(zb + d, p);
}

// h1[dst] += alpha * h[src]; one thread per (edge, 4-feature chunk)
__global__ void edge_aggregate(const int* __restrict__ src, const int* __restrict__ dst,
                               const float* __restrict__ ebuf, const float* __restrict__ zb,
                               const float* __restrict__ h, float* __restrict__ h1)
{
    const int idx = blockIdx.x * blockDim.x + threadIdx.x;
    const int e = idx >> 6;
    if (e >= E_EDGES) return;
    const int c = (idx & 63) << 2;
    const int s = src[e], d = dst[e];
    const float alpha = ebuf[e] / (zb[d] + 1e-16f);
    const float* hs = h  + (size_t)s * HID_F + c;
    float*       hd = h1 + (size_t)d * HID_F + c;
    atomicAdd(hd + 0, alpha * hs[0]);
    atomicAdd(hd + 1, alpha * hs[1]);
    atomicAdd(hd + 2, alpha * hs[2]);
    atomicAdd(hd + 3, alpha * hs[3]);
}

__global__ void elu_inplace(float* __restrict__ x, int n)
{
    const int i = blockIdx.x * blockDim.x + threadIdx.x;
    if (i < n) x[i] = elu_f(x[i]);
}

// ---------------- batch-norm (training stats, biased var) ----------------
__global__ void bn_stats(const float* __restrict__ z, float* __restrict__ mean,
                         float* __restrict__ var)
{
    __shared__ float ss[256], sq[256];
    const int c = blockIdx.x, t = threadIdx.x;
    float s = 0.f, q = 0.f;
    for (int r = t; r < N_NODES; r += 256) {
        const float v = z[(size_t)r * HID_F + c];
        s += v; q += v * v;
    }
    ss[t] = s; sq[t] = q;
    __syncthreads();
    for (int k = 128; k > 0; k >>= 1) {
        if (t < k) { ss[t] += ss[t + k]; sq[t] += sq[t + k]; }
        __syncthreads();
    }
    if (t == 0) {
        const float mu = ss[0] / (float)N_NODES;
        mean[c] = mu;
        var[c]  = sq[0] / (float)N_NODES - mu * mu;
    }
}

__global__ void bn_elu(float* __restrict__ z, const float* __restrict__ mean,
                       const float* __restrict__ var, const float* __restrict__ gamma,
                       const float* __restrict__ beta)
{
    const int i = blockIdx.x * blockDim.x + threadIdx.x;
    if (i >= N_NODES * HID_F) return;
    const int c = i & (HID_F - 1);
    const float v = (z[i] - mean[c]) * rsqrtf(var[c] + 1e-5f) * gamma[c] + beta[c];
    z[i] = elu_f(v);
}

// ---------------- readout ----------------
__global__ void mask_rowsum(const float* __restrict__ mask, float* __restrict__ rowsum)
{
    __shared__ float ss[256];
    const int r = blockIdx.x, t = threadIdx.x;
    float s = 0.f;
    for (int c = t; c < N_NODES; c += 256) s += mask[(size_t)r * N_NODES + c];
    ss[t] = s;
    __syncthreads();
    for (int k = 128; k > 0; k >>= 1) {
        if (t < k) ss[t] += ss[t + k];
        __syncthreads();
    }
    if (t == 0) rowsum[r] = ss[0];
}

__global__ void readout_g(const float* __restrict__ vsum, const float* __restrict__ rowsum,
                          float* __restrict__ g)
{
    __shared__ float ss[64];
    const int n = blockIdx.x, t = threadIdx.x;    // 64 threads
    const float v = vsum[(size_t)n * OUT_F + t] / rowsum[n];
    ss[t] = v * v;
    __syncthreads();
    for (int k = 32; k > 0; k >>= 1) {
        if (t < k) ss[t] += ss[t + k];
        __syncthreads();
    }
    const float nrm = fmaxf(sqrtf(ss[0]), 1e-12f);
    const float gv  = v / nrm;
    g[(size_t)n * OUT_F + t] = 1.f / (1.f + expf(-gv));
}

// ---------------- discriminator: out[n] = {hp[n].(W c[n]), hm[n].(W c[n])} + b
__global__ void disc_scores(const float* __restrict__ hp, const float* __restrict__ hm,
                            const float* __restrict__ c, const float* __restrict__ W,
                            const float* __restrict__ bptr, float* __restrict__ out2)
{
    __shared__ float s1[64], s2[64];
    const int n = blockIdx.x, d = threadIdx.x;    // 64 threads
    const float* cn = c + (size_t)n * OUT_F;
    float u = 0.f;
    for (int e = 0; e < OUT_F; ++e) u += W[d * OUT_F + e] * cn[e];
    s1[d] = hp[(size_t)n * OUT_F + d] * u;
    s2[d] = hm[(size_t)n * OUT_F + d] * u;
    __syncthreads();
    for (int k = 32; k > 0; k >>= 1) {
        if (d < k) { s1[d] += s1[d + k]; s2[d] += s2[d + k]; }
        __syncthreads();
    }
    if (d == 0) {
        const float b = bptr[0];
        out2[(size_t)n * 2 + 0] = s1[0] + b;
        out2[(size_t)n * 2 + 1] = s2[0] + b;
    }
}

__global__ void copy_f32(const float* __restrict__ s, float* __restrict__ d, int n)
{
    const int i = blockIdx.x * blockDim.x + threadIdx.x;
    if (i < n) d[i] = s[i];
}

// ---------------------------------------------------------------------------
extern "C" void kernel_launch(void* const* d_in, const int* in_sizes, int n_in,
                              void* d_out, int out_size, void* d_ws, size_t ws_size,
                              hipStream_t stream)
{
    (void)in_sizes; (void)n_in; (void)out_size; (void)ws_size;
    const float* feat   = (const float*)d_in[0];
    const float* feat_a = (const float*)d_in[1];
    const float* mask   = (const float*)d_in[2];
    const float* W1     = (const float*)d_in[3];
    const float* att_s  = (const float*)d_in[4];
    const float* att_d  = (const float*)d_in[5];
    const float* W2     = (const float*)d_in[6];
    const float* Wd1    = (const float*)d_in[7];
    const float* bd1    = (const float*)d_in[8];
    const float* gamma  = (const float*)d_in[9];
    const float* beta   = (const float*)d_in[10];
    const float* Wd2    = (const float*)d_in[11];
    const float* bd2    = (const float*)d_in[12];
    const float* discW  = (const float*)d_in[13];
    const float* discb  = (const float*)d_in[14];
    const int*   srcI   = (const int*)d_in[15];
    const int*   dstI   = srcI + E_EDGES;

    float* out = (float*)d_out;
    // output layout (flat concat in tuple order)
    const size_t O_H2   = 0;
    const size_t O_H3   = O_H2  + (size_t)N_NODES * OUT_F;
    const size_t O_RET  = O_H3  + (size_t)N_NODES * IN_F;
    const size_t O_RETA = O_RET + (size_t)N_NODES * 2;
    const size_t O_H2B  = O_RETA + (size_t)N_NODES * 2;
    const size_t O_H2A  = O_H2B + (size_t)N_NODES * OUT_F;

    // workspace carve-up (floats)
    float* w = (float*)d_ws;
    float* h_p  = w; w += (size_t)N_NODES * HID_F;
    float* h_m  = w; w += (size_t)N_NODES * HID_F;
    float* h1_p = w; w += (size_t)N_NODES * HID_F;
    float* h1_m = w; w += (size_t)N_NODES * HID_F;
    float* h2_p = w; w += (size_t)N_NODES * OUT_F;
    float* h2_m = w; w += (size_t)N_NODES * OUT_F;
    float* as_p = w; w += N_NODES;
    float* ad_p = w; w += N_NODES;
    float* as_m = w; w += N_NODES;
    float* ad_m = w; w += N_NODES;
    float* ebuf = w; w += E_EDGES;
    unsigned* mb = (unsigned*)w; w += N_NODES;
    float* zb   = w; w += N_NODES;
    float* meanb = w; w += HID_F;
    float* varb  = w; w += HID_F;
    float* rowsum = w; w += N_NODES;
    // reuse dead buffers
    float* zdec = h1_p;                 // alive after h2_p is formed
    float* vsum = h1_m;                 // alive after h2_m is formed
    float* g_p  = h_p;                  // h_p dead after aggregation P
    float* g_m  = h_p + (size_t)N_NODES * OUT_F;

    const dim3 B256(256);
    auto gemm = [&](const float* A, const float* Bm, const float* bias, float* C,
                    int M, int K, int Nc, int flags) {
        dim3 grid(M / 128, (Nc + 63) / 64);
        wmma_gemm_f32<<<grid, B256, 0, stream>>>(A, Bm, bias, C, M, K, Nc, flags);
    };

    // ---- encoder GEMMs: h = X @ W1
    gemm(feat,   W1, nullptr, h_p, N_NODES, IN_F, HID_F, 0);
    gemm(feat_a, W1, nullptr, h_m, N_NODES, IN_F, HID_F, 0);

    // ---- attention scalars
    att_dots<<<N_NODES, 256, 0, stream>>>(h_p, att_s, att_d, as_p, ad_p);
    att_dots<<<N_NODES, 256, 0, stream>>>(h_m, att_s, att_d, as_m, ad_m);

    const int initGrid = (N_NODES * HID_F) / 256;
    const int edgeGrid = E_EDGES / 256;
    const int aggGrid  = (E_EDGES * 64) / 256;

    // ---- graph-attention aggregation, feat path
    init_graph_bufs<<<initGrid, B256, 0, stream>>>(mb, zb, h1_p);
    edge_logits<<<edgeGrid, B256, 0, stream>>>(srcI, dstI, as_p, ad_p, ebuf, mb);
    edge_softmax_num<<<edgeGrid, B256, 0, stream>>>(dstI, mb, ebuf, zb);
    edge_aggregate<<<aggGrid, B256, 0, stream>>>(srcI, dstI, ebuf, zb, h_p, h1_p);
    elu_inplace<<<initGrid, B256, 0, stream>>>(h1_p, N_NODES * HID_F);
    gemm(h1_p, W2, nullptr, h2_p, N_NODES, HID_F, OUT_F, 0);

    // ---- graph-attention aggregation, feat_a path
    init_graph_bufs<<<initGrid, B256, 0, stream>>>(mb, zb, h1_m);
    edge_logits<<<edgeGrid, B256, 0, stream>>>(srcI, dstI, as_m, ad_m, ebuf, mb);
    edge_softmax_num<<<edgeGrid, B256, 0, stream>>>(dstI, mb, ebuf, zb);
    edge_aggregate<<<aggGrid, B256, 0, stream>>>(srcI, dstI, ebuf, zb, h_m, h1_m);
    elu_inplace<<<initGrid, B256, 0, stream>>>(h1_m, N_NODES * HID_F);
    gemm(h1_m, W2, nullptr, h2_m, N_NODES, HID_F, OUT_F, 0);

    // ---- decoder: z = h2@Wd1+bd1 ; BN ; elu ; h3 = z@Wd2+bd2
    gemm(h2_p, Wd1, bd1, zdec, N_NODES, OUT_F, HID_F, GF_BIAS);
    bn_stats<<<HID_F, B256, 0, stream>>>(zdec, meanb, varb);
    bn_elu<<<initGrid, B256, 0, stream>>>(zdec, meanb, varb, gamma, beta);
    gemm(zdec, Wd2, bd2, out + O_H3, N_NODES, HID_F, IN_F, GF_BIAS);

    // ---- readout: g = sigmoid(normalize((mask@h2)/rowsum))
    mask_rowsum<<<N_NODES, B256, 0, stream>>>(mask, rowsum);
    gemm(mask, h2_p, nullptr, vsum, N_NODES, N_NODES, OUT_F, 0);
    readout_g<<<N_NODES, 64, 0, stream>>>(vsum, rowsum, g_p);
    gemm(mask, h2_m, nullptr, vsum, N_NODES, N_NODES, OUT_F, 0);
    readout_g<<<N_NODES, 64, 0, stream>>>(vsum, rowsum, g_m);

    // ---- discriminator
    disc_scores<<<N_NODES, 64, 0, stream>>>(h2_p, h2_m, g_p, discW, discb, out + O_RET);
    disc_scores<<<N_NODES, 64, 0, stream>>>(h2_m, h2_p, g_m, discW, discb, out + O_RETA);

    // ---- h2 copies into output slots
    const int cpGrid = (N_NODES * OUT_F) / 256;
    copy_f32<<<cpGrid, B256, 0, stream>>>(h2_p, out + O_H2,  N_NODES * OUT_F);
    copy_f32<<<cpGrid, B256, 0, stream>>>(h2_p, out + O_H2B, N_NODES * OUT_F);
    copy_f32<<<cpGrid, B256, 0, stream>>>(h2_m, out + O_H2A, N_NODES * OUT_F);
}